// GaussianMomentDescriptor_26920855011348
// MI455X (gfx1250) — compile-verified
//
#include <hip/hip_runtime.h>
#include <hip/hip_bf16.h>

// ---------------------------------------------------------------------------
// GaussianMomentDescriptor for MI455X (gfx1250, wave32).
//
// Phase 1: per-neighbor radial/moment computation, symmetric-compact
//          scatter-add (L2-resident atomics). ~97% of pairs rejected by the
//          R_MAX cutoff before touching the embedding table or atomics.
// Phase 2: per-atom contractions. One atom per wave32; contr_1/2/3 come from
//          a single fp32 gram via chained V_WMMA_F32_16X16X4_F32, contr_6's
//          inner (ij)-contraction from a second WMMA gram. WMMA operands are
//          materialized zero-padded in LDS so operand fetch is a straight
//          ds_load_b64 per chunk (no divergent padding logic).
// ---------------------------------------------------------------------------

#define N_ATOMS   20000
#define N_NBRS    1200000
#define N_RADIAL  5
#define N_BASIS   7
#define N_SPECIES 119
#define R_MAXC    6.0f
#define R_MINC    0.5f

// compact per-atom moments: [atom][r][20] = 1 m0 + 3 m1 + 6 m2sym + 10 m3sym
#define CMOM_PER_ATOM (N_RADIAL * 20)
#define OUT_PER_ATOM  360

typedef float v2f __attribute__((ext_vector_type(2)));
typedef float v8f __attribute__((ext_vector_type(8)));

// (i,j) -> symmetric-pair slot, i,j in 0..2
__constant__ int c_s2map[9]  = {0,1,2, 1,3,4, 2,4,5};
// (i,j,k) flat (i*9+j*3+k) -> symmetric-triple slot
__constant__ int c_s3map[27] = {0,1,2, 1,3,4, 2,4,5,
                                1,3,4, 3,6,7, 4,7,8,
                                2,4,5, 4,7,8, 5,8,9};
// tril_2d_indices(5): 15 pairs in reference order
__constant__ int c_p2i[15] = {0,0,0,0,0, 1,1,1,1, 2,2,2, 3,3, 4};
__constant__ int c_p2j[15] = {0,1,2,3,4, 1,2,3,4, 2,3,4, 3,4, 4};
// tril_3d_indices(5): 35 triples in reference order
__constant__ int c_t3[35][3] = {
  {0,0,0},{0,0,1},{0,0,2},{0,0,3},{0,0,4},
  {0,1,1},{0,1,2},{0,1,3},{0,1,4},
  {0,2,2},{0,2,3},{0,2,4},
  {0,3,3},{0,3,4},
  {0,4,4},
  {1,1,1},{1,1,2},{1,1,3},{1,1,4},
  {1,2,2},{1,2,3},{1,2,4},
  {1,3,3},{1,3,4},
  {1,4,4},
  {2,2,2},{2,2,3},{2,2,4},
  {2,3,3},{2,3,4},
  {2,4,4},
  {3,3,3},{3,3,4},
  {3,4,4},
  {4,4,4}
};

// ------------------------------- zero init ---------------------------------
__global__ void gm_zero_kernel(float* __restrict__ p, int n) {
  int t = blockIdx.x * blockDim.x + threadIdx.x;
  if (t < n) p[t] = 0.0f;
}

// --------------------------- phase 1: neighbors ----------------------------
__global__ __launch_bounds__(256)
void gm_neighbor_kernel(const float* __restrict__ R, const int* __restrict__ Z,
                        const int* __restrict__ nbr, const float* __restrict__ emb,
                        float* __restrict__ mom) {
  int n = blockIdx.x * blockDim.x + threadIdx.x;
  if (n >= N_NBRS) return;
  int i = nbr[n];
  int j = nbr[N_NBRS + n];
  if (i == j) return;  // reference zeroes self-pairs

  float xi = R[3*i], yi = R[3*i+1], zi = R[3*i+2];
  float xj = R[3*j], yj = R[3*j+1], zj = R[3*j+2];
  float dx = xj - xi, dy = yj - yi, dz = zj - zi;
  float dr = sqrtf(dx*dx + dy*dy + dz*dz);
  if (dr >= R_MAXC) return;  // cutoff = 0 -> contributes nothing

  // gather embedding row early and prefetch it behind the transcendental work
  const float* co = emb + (size_t)(Z[j] * N_SPECIES + Z[i]) * (N_RADIAL * N_BASIS);
  __builtin_prefetch(co, 0, 0);  // -> global_prefetch_b8

  float inv = 1.0f / (dr + 1e-5f);
  float d0 = dx * inv, d1 = dy * inv, d2 = dz * inv;

  const float kPi = 3.14159265358979f;
  float cut = 0.5f * (__cosf(kPi * dr * (1.0f / R_MAXC)) + 1.0f);
  const float betta = (float)(N_BASIS * N_BASIS) / (R_MAXC * R_MAXC);  // 49/36
  const float rad_norm = 0.96481514f;    // (2*betta/pi)^0.25
  const float inv_sqrt_nb = 0.37796447f; // 1/sqrt(7)
  float scale = cut * inv_sqrt_nb;

  float basis[N_BASIS];
#pragma unroll
  for (int b = 0; b < N_BASIS; ++b) {
    float sh = (R_MINC + ((R_MAXC - R_MINC) / N_BASIS) * (float)b) - dr;
    basis[b] = rad_norm * __expf(-betta * sh * sh);
  }

  // symmetric products of the unit vector
  float dd[6]   = {d0*d0, d0*d1, d0*d2, d1*d1, d1*d2, d2*d2};
  float ddd[10] = {dd[0]*d0, dd[0]*d1, dd[0]*d2,
                   dd[3]*d0, dd[1]*d2, dd[5]*d0,
                   dd[3]*d1, dd[3]*d2, dd[5]*d1, dd[5]*d2};

  float* base = mom + (size_t)j * CMOM_PER_ATOM;
#pragma unroll
  for (int r = 0; r < N_RADIAL; ++r) {
    float rad = 0.0f;
#pragma unroll
    for (int b = 0; b < N_BASIS; ++b) rad = fmaf(co[r * N_BASIS + b], basis[b], rad);
    rad *= scale;
    float* p = base + r * 20;
    atomicAdd(p + 0, rad);
    atomicAdd(p + 1, rad * d0);
    atomicAdd(p + 2, rad * d1);
    atomicAdd(p + 3, rad * d2);
#pragma unroll
    for (int q = 0; q < 6; ++q)  atomicAdd(p + 4 + q,  rad * dd[q]);
#pragma unroll
    for (int q = 0; q < 10; ++q) atomicAdd(p + 10 + q, rad * ddd[q]);
  }
}

// ----------------------- phase 2: per-atom contractions --------------------
// full-form LDS layout per atom: m0 at [0..5), m1 at [5 + r*3 + i],
// m2 at [20 + r*9 + i*3 + j], m3 at [65 + r*27 + i*9 + j*3 + k]
//
// X (16 rows x 28 K, zero padded): rows 0-4 = m1[r], 5-9 = m2[r], 10-14 = m3[r]
// W (16 rows x 12 K, zero padded): W[kk*5+r][ij] = m3[r, ij, kk]

#define XSTRIDE 28
#define WSTRIDE 12

__global__ __launch_bounds__(256)
void gm_contract_kernel(const float* __restrict__ mom, float* __restrict__ out) {
  __shared__ float sM[8][200];            // expanded moments per wave/atom
  __shared__ float sX[8][16 * XSTRIDE];   // padded gram operand
  __shared__ float sW[8][16 * WSTRIDE];   // padded contr_6 operand
  __shared__ float sG[8][256];            // 16x16 gram D (contr_1/2/3 blocks)
  __shared__ float sQ[8][256];            // 16x16 Q matrix for contr_6

  int lane = threadIdx.x & 31;
  int w    = threadIdx.x >> 5;
  int atom = blockIdx.x * 8 + w;  // grid = 2500 -> exactly 20000, full waves

  const float* cm = mom + (size_t)atom * CMOM_PER_ATOM;
  float* m = sM[w];
  float* X = sX[w];
  float* W = sW[w];

  // zero the padded operand matrices (per-wave region; in-order DS within wave)
  for (int t = lane; t < 16 * XSTRIDE; t += 32) X[t] = 0.0f;
  for (int t = lane; t < 16 * WSTRIDE; t += 32) W[t] = 0.0f;

  // expand symmetric-compact -> full form + padded operand matrices
  for (int f = lane; f < 200; f += 32) {
    int slot;
    if (f < 5)        slot = f * 20;
    else if (f < 20)  { int g = f - 5;  slot = (g / 3)  * 20 + 1  + (g % 3); }
    else if (f < 65)  { int g = f - 20; slot = (g / 9)  * 20 + 4  + c_s2map[g % 9]; }
    else              { int g = f - 65; slot = (g / 27) * 20 + 10 + c_s3map[g % 27]; }
    float v = cm[slot];
    m[f] = v;
    if (f >= 5 && f < 20) {
      int g = f - 5;  X[(g / 3) * XSTRIDE + (g % 3)] = v;
    } else if (f >= 20 && f < 65) {
      int g = f - 20; X[(5 + g / 9) * XSTRIDE + (g % 9)] = v;
    } else if (f >= 65) {
      int g = f - 65; int r = g / 27, c = g % 27;
      X[(10 + r) * XSTRIDE + c] = v;
      W[((c % 3) * 5 + r) * WSTRIDE + (c / 3)] = v;  // ij = c/3, kk = c%3
    }
  }
  __syncthreads();

  // Per-lane operand coords for V_WMMA_F32_16X16X4_F32:
  //   A (16x4): lanes 0-15 -> M=lane, VGPR{0,1}=K{0,1}; lanes 16-31 -> K{2,3}
  //   B (4x16): mirrored layout, so for a symmetric gram B's lane data == A's.
  int Mrow = lane & 15;
  int kOff = (lane >> 4) << 1;  // 0 or 2
  const float* xrow = X + Mrow * XSTRIDE + kOff;
  const float* wrow = W + Mrow * WSTRIDE + kOff;

  // Gram G = X * X^T  (K = 28, 7 chained K=4 WMMAs); each operand = 1 ds_load_b64
  v8f acc = {};
#pragma unroll
  for (int kc = 0; kc < 7; ++kc) {
    v2f a = *(const v2f*)(xrow + kc * 4);
    acc = __builtin_amdgcn_wmma_f32_16x16x4_f32(false, a, false, a,
                                                (short)0, acc, false, false);
  }
  // D layout: VGPR v, lanes 0-15 -> (M=v, N=lane); lanes 16-31 -> (M=v+8, N=lane-16)
  float* G = sG[w];
#pragma unroll
  for (int v = 0; v < 8; ++v)
    G[(v + 8 * (lane >> 4)) * 16 + (lane & 15)] = acc[v];

  // Q = W * W^T  (K = 12, 3 WMMAs): Q[(k*5+r),(l*5+s)] = sum_ij m3[r,ij,k]*m3[s,ij,l]
  v8f accq = {};
#pragma unroll
  for (int kc = 0; kc < 3; ++kc) {
    v2f a = *(const v2f*)(wrow + kc * 4);
    accq = __builtin_amdgcn_wmma_f32_16x16x4_f32(false, a, false, a,
                                                 (short)0, accq, false, false);
  }
  float* Q = sQ[w];
#pragma unroll
  for (int v = 0; v < 8; ++v)
    Q[(v + 8 * (lane >> 4)) * 16 + (lane & 15)] = accq[v];
  __syncthreads();

  float* o = out + (size_t)atom * OUT_PER_ATOM;
  const float* m0 = m;
  const float* m1 = m + 5;
  const float* m2 = m + 20;
  const float* m3 = m + 65;

  // contr_0 : [0..5)
  if (lane < 5) o[lane] = m0[lane];

  // contr_1/2/3 : [5..50) from the diagonal 5x5 blocks of G
  for (int p = lane; p < 15; p += 32) {
    int i2 = c_p2i[p], j2 = c_p2j[p];
    o[5  + p] = G[i2 * 16 + j2];
    o[20 + p] = G[(5 + i2) * 16 + (5 + j2)];
    o[35 + p] = G[(10 + i2) * 16 + (10 + j2)];
  }

  // contr_4 : [50..85)  sum_ijk m2[r,i,j] m2[s,i,k] m2[t,j,k]
  for (int q = lane; q < 35; q += 32) {
    int r = c_t3[q][0], s = c_t3[q][1], t = c_t3[q][2];
    float a4 = 0.0f;
    for (int i = 0; i < 3; ++i)
      for (int jj = 0; jj < 3; ++jj)
        for (int k = 0; k < 3; ++k)
          a4 = fmaf(m2[r*9 + i*3 + jj] * m2[s*9 + i*3 + k], m2[t*9 + jj*3 + k], a4);
    o[50 + q] = a4;
  }

  // contr_5 : [85..160)  sum_ij m1[r,i] m1[s,j] m2[t,i,j]
  for (int e = lane; e < 75; e += 32) {
    int p = e / 5, t = e % 5;
    int r = c_p2i[p], s = c_p2j[p];
    float a5 = 0.0f;
    for (int i = 0; i < 3; ++i)
      for (int jj = 0; jj < 3; ++jj)
        a5 = fmaf(m1[r*3 + i] * m1[s*3 + jj], m2[t*9 + i*3 + jj], a5);
    o[85 + e] = a5;
  }

  // contr_6 : [160..235)  sum_kl Q[(k,r),(l,s)] * m2[t,k,l]
  for (int e = lane; e < 75; e += 32) {
    int p = e / 5, t = e % 5;
    int r = c_p2i[p], s = c_p2j[p];
    float a6 = 0.0f;
    for (int k = 0; k < 3; ++k)
      for (int l = 0; l < 3; ++l)
        a6 = fmaf(Q[(k*5 + r) * 16 + (l*5 + s)], m2[t*9 + k*3 + l], a6);
    o[160 + e] = a6;
  }

  // contr_7 : [235..360)  sum_ijk m3[r,i,j,k] m2[s,i,j] m1[t,k]
  for (int e = lane; e < 125; e += 32) {
    int r = e / 25, s = (e / 5) % 5, t = e % 5;
    float a7 = 0.0f;
    for (int i = 0; i < 3; ++i)
      for (int jj = 0; jj < 3; ++jj)
        for (int k = 0; k < 3; ++k)
          a7 = fmaf(m3[r*27 + i*9 + jj*3 + k] * m2[s*9 + i*3 + jj], m1[t*3 + k], a7);
    o[235 + e] = a7;
  }
}

// ------------------------------- launcher ----------------------------------
extern "C" void kernel_launch(void* const* d_in, const int* in_sizes, int n_in,
                              void* d_out, int out_size, void* d_ws, size_t ws_size,
                              hipStream_t stream) {
  (void)in_sizes; (void)n_in; (void)out_size; (void)ws_size;
  const float* R   = (const float*)d_in[0];
  const int*   Z   = (const int*)d_in[1];
  const int*   nbr = (const int*)d_in[2];
  const float* emb = (const float*)d_in[3];
  // d_in[4] = box (unused: reference ignores it)

  float* mom = (float*)d_ws;  // 20000 * 100 floats = 8 MB (L2-resident)
  int nmom = N_ATOMS * CMOM_PER_ATOM;

  gm_zero_kernel<<<(nmom + 255) / 256, 256, 0, stream>>>(mom, nmom);
  gm_neighbor_kernel<<<(N_NBRS + 255) / 256, 256, 0, stream>>>(R, Z, nbr, emb, mom);
  gm_contract_kernel<<<N_ATOMS / 8, 256, 0, stream>>>(mom, (float*)d_out);
}